// HarmonicLNNModel_60138132079194
// MI455X (gfx1250) — compile-verified
//
#include <hip/hip_runtime.h>
#include <math.h>

// ---------------- problem constants ----------------
#define HID     64
#define NIN     5
#define NAPP    4
#define NHEADS  5
#define NOUTC   (NHEADS * NAPP)   // 20 head columns
#define TSTEPS  2048
#define NBATCH  4096
#define DTC     0.1f
#define EPSC    1e-5f

// augmented K: 64 (h) + 5 (x) + 1 (bias one) + 2 pad = 72 -> 18 chunks of K=4
#define KAUG    72
#define NCHUNKS 18
#define SSTRIDE 74                // padded row stride of A-staging tile (floats)

typedef float v2f __attribute__((ext_vector_type(2)));
typedef float v8f __attribute__((ext_vector_type(8)));

#if __has_builtin(__builtin_amdgcn_wmma_f32_16x16x4_f32)
#define HAVE_WMMA_F32X4 1
#else
#define HAVE_WMMA_F32X4 0
#endif

#if __has_builtin(__builtin_amdgcn_permlane16)
#define HAVE_PERMLANE16 1
#else
#define HAVE_PERMLANE16 0
#endif

__device__ __forceinline__ float tanh_fast(float x) {
#if __has_builtin(__builtin_amdgcn_tanhf)
  return __builtin_amdgcn_tanhf(x);
#elif __has_builtin(__builtin_amdgcn_tanh_f32)
  return __builtin_amdgcn_tanh_f32(x);
#else
  return tanhf(x);
#endif
}

__device__ __forceinline__ float rsqrt_fast(float x) {
#if __has_builtin(__builtin_amdgcn_rsqf)
  return __builtin_amdgcn_rsqf(x);
#else
  return rsqrtf(x);
#endif
}

// xor-butterfly add within 16-lane groups (both wave halves independently).
// permlane16 = pure VALU cross-lane (no DS round trip, no DScnt waits).
template <int M, unsigned S0, unsigned S1>
__device__ __forceinline__ float bfly_add(float v) {
#if HAVE_PERMLANE16
  const unsigned u = (unsigned)__float_as_int(v);
  const unsigned p = __builtin_amdgcn_permlane16(u, u, S0, S1, false, false);
  return v + __int_as_float((int)p);
#else
  return v + __shfl_xor(v, M);
#endif
}

__device__ __forceinline__ float bfly_sum16(float v) {
  v = bfly_add<1, 0x67452301u, 0xEFCDAB89u>(v);  // sel[i] = i^1
  v = bfly_add<2, 0x54761032u, 0xDCFE98BAu>(v);  // sel[i] = i^2
  v = bfly_add<4, 0x32107654u, 0xBA98FEDCu>(v);  // sel[i] = i^4
  v = bfly_add<8, 0xFEDCBA98u, 0x76543210u>(v);  // sel[i] = i^8
  return v;
}

// LDS (floats): wT 64x72 (transposed augmented weights, kept resident)
//               stag 16xSSTRIDE : [ h(64) | x(5) | 1 | pad(0,0) ... ]
#define SM_WT    0
#define SM_STAG  (HID * KAUG)          // 4608
#define SM_TOTAL (SM_STAG + 16 * SSTRIDE)

__global__ __launch_bounds__(32, 1)
void hlnn_fused_kernel(const float* __restrict__ x,
                       const float* __restrict__ W_in,
                       const float* __restrict__ b_in,
                       const float* __restrict__ tau_param,
                       const float* __restrict__ W_rec,
                       const float* __restrict__ ln1_w,
                       const float* __restrict__ ln1_b,
                       const float* __restrict__ ln2_w,
                       const float* __restrict__ ln2_b,
                       const float* __restrict__ head_W,
                       const float* __restrict__ head_b,
                       float* __restrict__ out)
{
  __shared__ __align__(16) float smem[SM_TOTAL];
  float* wT   = smem + SM_WT;
  float* stag = smem + SM_STAG;

  const int lane = threadIdx.x;        // blockDim == 32: one wave per WG
  const int l16  = lane & 15;
  const int hi   = lane >> 4;          // 0: lanes 0-15, 1: lanes 16-31
  const int b0   = blockIdx.x * 16;    // batch-row tile base

  // ---- init: augmented W^T (64 x 72) into LDS, zero A-staging tile ----
  for (int idx = lane; idx < HID * KAUG; idx += 32) {
    const int n = idx / KAUG;
    const int k = idx - n * KAUG;
    float v;
    if      (k < HID)  v = W_rec[k * HID + n];          // h @ W_rec
    else if (k < 69)   v = W_in[(k - HID) * HID + n];   // x @ W_in
    else if (k == 69)  v = b_in[n];                     // + b_in via "1" column
    else               v = 0.0f;                        // pad
    wT[idx] = v;
  }
  for (int idx = lane; idx < 16 * SSTRIDE; idx += 32) stag[idx] = 0.0f;   // h0 = 0
  __syncthreads();
  if (lane < 16) stag[lane * SSTRIDE + 69] = 1.0f;      // constant "1" column

  // per-lane column constants for the 4 N-tiles this wave owns
  float w1c[4], b1c[4], w2c[4], b2c[4], hdecay[4];
  #pragma unroll
  for (int n = 0; n < 4; ++n) {
    const int c = 16 * n + l16;
    w1c[n] = ln1_w[c];  b1c[n] = ln1_b[c];
    w2c[n] = ln2_w[c];  b2c[n] = ln2_b[c];
    const float inv_tau = 1.0f / log1pf(expf(tau_param[c]));   // 1/softplus
    hdecay[n] = 1.0f - DTC * inv_tau;                          // h*(1-dt/tau)+dt*f
  }

  // persistent B operands: 18 K-chunks x 4 N-tiles, 2 floats/lane each (144 VGPRs)
  v2f breg[4][NCHUNKS];
  #pragma unroll
  for (int n = 0; n < 4; ++n) {
    const float* bp = wT + (16 * n + l16) * KAUG + 2 * hi;
    #pragma unroll
    for (int c = 0; c < NCHUNKS; ++c)
      breg[n][c] = *(const v2f*)(bp + 4 * c);
  }
  __syncthreads();

  // stage x(t=0) into A-tile columns 64..68
  {
    const float* xr = x + (size_t)(b0 + l16) * TSTEPS * NIN;
    if (hi == 0) {
      stag[l16 * SSTRIDE + 64] = xr[0];
      stag[l16 * SSTRIDE + 65] = xr[1];
      stag[l16 * SSTRIDE + 66] = xr[2];
    } else {
      stag[l16 * SSTRIDE + 67] = xr[3];
      stag[l16 * SSTRIDE + 68] = xr[4];
    }
  }

  // persistent hidden state, C/D layout: hreg[n][v] = element (row v+8*hi, col 16n+l16)
  float hreg[4][8];
  #pragma unroll
  for (int n = 0; n < 4; ++n)
    #pragma unroll
    for (int v = 0; v < 8; ++v) hreg[n][v] = 0.0f;

  // ---------------- recurrence: 2048 steps, zero barriers ----------------
  for (int t = 0; t < TSTEPS; ++t) {
    // register-prefetch x(t+1) (x fits in 192MB L2; consecutive t contiguous per row)
    const int tl = (t + 1 < TSTEPS) ? t + 1 : TSTEPS - 1;
    const float* xr = x + ((size_t)(b0 + l16) * TSTEPS + tl) * NIN;
    float xn0, xn1, xn2;
    if (hi == 0) { xn0 = xr[0]; xn1 = xr[1]; xn2 = xr[2]; }
    else         { xn0 = xr[3]; xn1 = xr[4]; xn2 = 0.0f; }
    if ((t & 3) == 0) {      // warm L2 from HBM ~16 steps ahead
      const int tp = (t + 16 < TSTEPS) ? t + 16 : TSTEPS - 1;
      if (hi == 0)
        __builtin_prefetch(x + ((size_t)(b0 + l16) * TSTEPS + tp) * NIN, 0, 3);
    }

    // z = [h | x_t | 1] @ Waug : 16x72 * 72x64 as 4 independent 18-deep WMMA chains
    v8f acc[4];
    #pragma unroll
    for (int n = 0; n < 4; ++n) {
      const v8f z = {0.f, 0.f, 0.f, 0.f, 0.f, 0.f, 0.f, 0.f};
      acc[n] = z;
    }
    const float* ap = stag + l16 * SSTRIDE + 2 * hi;
#if HAVE_WMMA_F32X4
    #pragma unroll
    for (int c = 0; c < NCHUNKS; ++c) {
      const v2f a = *(const v2f*)(ap + 4 * c);          // shared A operand
      #pragma unroll
      for (int n = 0; n < 4; ++n)
        acc[n] = __builtin_amdgcn_wmma_f32_16x16x4_f32(false, a, false, breg[n][c],
                                                       (short)0, acc[n], false, false);
    }
#else
    (void)ap;
    #pragma unroll
    for (int n = 0; n < 4; ++n)
      #pragma unroll
      for (int v = 0; v < 8; ++v) {
        const int row = v + 8 * hi, col = 16 * n + l16;
        float s = 0.0f;
        for (int k = 0; k < KAUG; ++k) s += stag[row * SSTRIDE + k] * wT[col * KAUG + k];
        acc[n][v] = s;
      }
#endif

    // LayerNorm + tanh + state update. Row v+8*hi's full 64-col sum = 4-tile partial
    // + 16-lane VALU butterfly; every lane ends holding exactly its own row's stats.
    #pragma unroll
    for (int v = 0; v < 8; ++v) {
      const float z0 = acc[0][v], z1 = acc[1][v], z2 = acc[2][v], z3 = acc[3][v];
      float s = (z0 + z1) + (z2 + z3);
      float q = (z0 * z0 + z1 * z1) + (z2 * z2 + z3 * z3);
      s = bfly_sum16(s);
      q = bfly_sum16(q);
      const float mu  = s * (1.0f / 64.0f);
      const float var = q * (1.0f / 64.0f) - mu * mu;
      const float rs  = rsqrt_fast(var + EPSC);
      const int row = v + 8 * hi;
      #pragma unroll
      for (int n = 0; n < 4; ++n) {
        const float t1 = rs * w1c[n];
        const float f  = tanh_fast((acc[n][v] - mu) * t1 + b1c[n]);
        float hn = hreg[n][v] * hdecay[n] + DTC * f;
        hn = fminf(fmaxf(hn, -10.0f), 10.0f);
        hreg[n][v] = hn;
        stag[row * SSTRIDE + 16 * n + l16] = hn;   // rebuild A tile (same-wave LDS: in order)
      }
    }

    // stage x(t+1) into A-tile columns 64..68
    if (hi == 0) {
      stag[l16 * SSTRIDE + 64] = xn0;
      stag[l16 * SSTRIDE + 65] = xn1;
      stag[l16 * SSTRIDE + 66] = xn2;
    } else {
      stag[l16 * SSTRIDE + 67] = xn0;
      stag[l16 * SSTRIDE + 68] = xn1;
    }
  }

  // ---------------- final LayerNorm (ln2) on h_T -> stag cols 0..63 ----------------
  #pragma unroll
  for (int v = 0; v < 8; ++v) {
    const float z0 = hreg[0][v], z1 = hreg[1][v], z2 = hreg[2][v], z3 = hreg[3][v];
    float s = (z0 + z1) + (z2 + z3);
    float q = (z0 * z0 + z1 * z1) + (z2 * z2 + z3 * z3);
    s = bfly_sum16(s);
    q = bfly_sum16(q);
    const float mu  = s * (1.0f / 64.0f);
    const float var = q * (1.0f / 64.0f) - mu * mu;
    const float rs  = rsqrt_fast(var + EPSC);
    const int row = v + 8 * hi;
    #pragma unroll
    for (int n = 0; n < 4; ++n)
      stag[row * SSTRIDE + 16 * n + l16] = (hreg[n][v] - mu) * rs * w2c[n] + b2c[n];
  }

  // ---------------- heads via WMMA: (16x64) @ (64x20, padded to 32) ----------------
  v8f oacc[2];
  #pragma unroll
  for (int m = 0; m < 2; ++m) {
    const v8f z = {0.f, 0.f, 0.f, 0.f, 0.f, 0.f, 0.f, 0.f};
    oacc[m] = z;
  }
#if HAVE_WMMA_F32X4
  {
    const float* ap = stag + l16 * SSTRIDE + 2 * hi;
    #pragma unroll
    for (int c = 0; c < HID / 4; ++c) {
      const v2f a = *(const v2f*)(ap + 4 * c);
      #pragma unroll
      for (int m = 0; m < 2; ++m) {
        const int col = 16 * m + l16;
        v2f b = {0.0f, 0.0f};
        if (col < NOUTC) b = *(const v2f*)(head_W + col * HID + 4 * c + 2 * hi);
        oacc[m] = __builtin_amdgcn_wmma_f32_16x16x4_f32(false, a, false, b,
                                                        (short)0, oacc[m], false, false);
      }
    }
  }
#else
  #pragma unroll
  for (int m = 0; m < 2; ++m) {
    const int col = 16 * m + l16;
    if (col < NOUTC)
      #pragma unroll
      for (int v = 0; v < 8; ++v) {
        const int row = v + 8 * hi;
        float s = 0.0f;
        for (int j = 0; j < HID; ++j) s += stag[row * SSTRIDE + j] * head_W[col * HID + j];
        oacc[m][v] = s;
      }
  }
#endif
  #pragma unroll
  for (int m = 0; m < 2; ++m) {
    const int col = 16 * m + l16;
    if (col < NOUTC) {
      const float hb = head_b[col];
      const int k = col >> 2, a = col & 3;
      #pragma unroll
      for (int v = 0; v < 8; ++v) {
        const int row = v + 8 * hi;
        out[((size_t)k * NBATCH + b0 + row) * NAPP + a] = oacc[m][v] + hb;
      }
    }
  }
}

extern "C" void kernel_launch(void* const* d_in, const int* in_sizes, int n_in,
                              void* d_out, int out_size, void* d_ws, size_t ws_size,
                              hipStream_t stream) {
  (void)in_sizes; (void)n_in; (void)out_size; (void)d_ws; (void)ws_size;
  const float* x     = (const float*)d_in[0];
  const float* W_in  = (const float*)d_in[1];
  const float* b_in  = (const float*)d_in[2];
  const float* taup  = (const float*)d_in[3];
  const float* W_rec = (const float*)d_in[4];
  const float* l1w   = (const float*)d_in[5];
  const float* l1b   = (const float*)d_in[6];
  const float* l2w   = (const float*)d_in[7];
  const float* l2b   = (const float*)d_in[8];
  const float* hW    = (const float*)d_in[9];
  const float* hb    = (const float*)d_in[10];
  float* out = (float*)d_out;

  dim3 grid(NBATCH / 16);
  dim3 block(32);
  hlnn_fused_kernel<<<grid, block, 0, stream>>>(x, W_in, b_in, taup, W_rec,
                                                l1w, l1b, l2w, l2b, hW, hb, out);
}